// DetectionLoss_68109591380483
// MI455X (gfx1250) — compile-verified
//
#include <hip/hip_runtime.h>
#include <hip/hip_bf16.h>
#include <stdint.h>

// ---------------------------------------------------------------------------
// DetectionLoss for MI455X (gfx1250, wave32).
// Memory-bound streaming pass + exact radix-select hard-negative mining +
// CDNA5 TDM gt-box prefetch + f32 WMMA final reduction.
// ---------------------------------------------------------------------------

#define WGSZ 256
constexpr int   B_IMG  = 64;
constexpr int   G_GT   = 32;
constexpr float F_EPS  = 1e-6f;

typedef float        v2f   __attribute__((ext_vector_type(2)));
typedef float        v8f   __attribute__((ext_vector_type(8)));
typedef unsigned int u32x4 __attribute__((ext_vector_type(4)));
typedef int          i32x8 __attribute__((ext_vector_type(8)));
typedef int          i32x4 __attribute__((ext_vector_type(4)));

#if defined(__has_builtin)
#  if __has_builtin(__builtin_amdgcn_tensor_load_to_lds) && __has_builtin(__builtin_amdgcn_s_wait_tensorcnt)
#    define HAVE_TDM 1
#  else
#    define HAVE_TDM 0
#  endif
#  if __has_builtin(__builtin_amdgcn_wmma_f32_16x16x4_f32)
#    define HAVE_WMMA_F32 1
#  else
#    define HAVE_WMMA_F32 0
#  endif
#else
#  define HAVE_TDM 0
#  define HAVE_WMMA_F32 0
#endif

__device__ __forceinline__ float waveSumF(float v) {
    #pragma unroll
    for (int o = 16; o > 0; o >>= 1) v += __shfl_xor(v, o, 32);
    return v;
}
__device__ __forceinline__ int waveSumI(int v) {
    #pragma unroll
    for (int o = 16; o > 0; o >>= 1) v += __shfl_xor(v, o, 32);
    return v;
}
__device__ __forceinline__ float sl1(float x) {
    float ax = fabsf(x);
    return (ax < 1.0f) ? 0.5f * x * x : ax - 0.5f;
}

// --------------------------- init ------------------------------------------
__global__ void dl_init(float* partF, int* partI) {
    int t = threadIdx.x;
    for (int i = t; i < 192 * 3; i += WGSZ) partF[i] = 0.0f;
    for (int i = t; i < 192 * 2; i += WGSZ) partI[i] = 0;
}

// --------------------------- pass A ----------------------------------------
// grid = (HW/256, 64); thread owns one spatial site (3 anchors).
__launch_bounds__(WGSZ)
__global__ void dl_pass_a(const float* __restrict__ pred,
                          const float* __restrict__ anchors,
                          const float* __restrict__ gt_boxes,
                          const int*   __restrict__ gt_labels,
                          int HW,
                          uint32_t* __restrict__ keys,   // per-scale base, [B][HW*3]
                          float* __restrict__ partF,     // [192*3] loc/obj_pos/cls
                          int*   __restrict__ partI,     // [192*2] pos/neg counts
                          int scale) {
    __shared__ float s_box[G_GT][4];
    __shared__ float s_area[G_GT], s_gx[G_GT], s_gy[G_GT], s_gw[G_GT], s_gh[G_GT];
    __shared__ int   s_lab[G_GT];
    __shared__ float s_redF[3][8];
    __shared__ int   s_redI[2][8];

    const int tid  = threadIdx.x;
    const int b    = blockIdx.y;
    const int wave = tid >> 5;
    const int lane = tid & 31;

    if (wave == 0) {
#if HAVE_TDM
        // Tensor Data Mover: DMA 32 GT boxes (1x128 f32 tile) into LDS.
        uint64_t ga      = (uint64_t)(uintptr_t)(gt_boxes + (size_t)b * G_GT * 4);
        uint32_t lds_off = (uint32_t)(uintptr_t)(&s_box[0][0]);  // flat low 32 = LDS offset
        u32x4 g0; i32x8 g1;
        i32x4 g2 = {0, 0, 0, 0}, g3 = {0, 0, 0, 0};
        i32x8 g4 = {0, 0, 0, 0, 0, 0, 0, 0};
        g0[0] = 1u;                                                // count = 1 (valid)
        g0[1] = lds_off;                                           // lds_addr
        g0[2] = (uint32_t)ga;                                      // global_addr[31:0]
        g0[3] = (uint32_t)((ga >> 32) & 0x1FFFFFFu) | 0x80000000u; // addr[56:32] | type=2
        g1[0] = 2 << 16;          // data_size = 4B; no mask/iterate/pad
        g1[1] = 128 << 16;        // tensor_dim0 = 128 (lo16)
        g1[2] = 1 << 16;          // tensor_dim0 hi=0 | tensor_dim1 = 1 (lo16)
        g1[3] = 128 << 16;        // tensor_dim1 hi=0 | tile_dim0 = 128
        g1[4] = 1;                // tile_dim1 = 1, tile_dim2 = 0
        g1[5] = 128;              // tensor_dim0_stride = 128
        g1[6] = 0;
        g1[7] = 0;
        __builtin_amdgcn_tensor_load_to_lds(g0, g1, g2, g3, g4, 0);
        __builtin_amdgcn_s_wait_tensorcnt(0);
#else
        if (lane < G_GT) {
            const float4 bb = ((const float4*)gt_boxes)[b * G_GT + lane];
            s_box[lane][0] = bb.x; s_box[lane][1] = bb.y;
            s_box[lane][2] = bb.z; s_box[lane][3] = bb.w;
        }
#endif
        if (lane < G_GT) {
            float x1 = s_box[lane][0], y1 = s_box[lane][1];
            float x2 = s_box[lane][2], y2 = s_box[lane][3];
            s_area[lane] = (x2 - x1) * (y2 - y1);
            s_gx[lane]   = (x1 + x2) * 0.5f;
            s_gy[lane]   = (y1 + y2) * 0.5f;
            s_gw[lane]   = fmaxf(x2 - x1, F_EPS);
            s_gh[lane]   = fmaxf(y2 - y1, F_EPS);
            s_lab[lane]  = gt_labels[b * G_GT + lane];
        }
    }
    __syncthreads();

    const int    hw       = blockIdx.x * WGSZ + tid;       // HW is a multiple of 256
    const size_t predBase = (size_t)b * 24 * HW + hw;

    float t_loc = 0.f, t_obj = 0.f, t_cls = 0.f;
    int   t_pos = 0, t_neg = 0;

    #pragma unroll
    for (int a = 0; a < 3; ++a) {
        float p[8];
        #pragma unroll
        for (int c = 0; c < 8; ++c)
            p[c] = pred[predBase + (size_t)(a * 8 + c) * HW];   // coalesced per plane

        const int n = hw * 3 + a;
        float4 anc = ((const float4*)anchors)[n];
        const float ax1 = anc.x, ay1 = anc.y, ax2 = anc.z, ay2 = anc.w;
        const float areaA = (ax2 - ax1) * (ay2 - ay1);

        float best = -1.0f; int bi = 0;
        #pragma unroll 4
        for (int g = 0; g < G_GT; ++g) {
            float ix = fmaxf(fminf(ax2, s_box[g][2]) - fmaxf(ax1, s_box[g][0]), 0.f);
            float iy = fmaxf(fminf(ay2, s_box[g][3]) - fmaxf(ay1, s_box[g][1]), 0.f);
            float inter = ix * iy;
            float iou = inter / (areaA + s_area[g] - inter + 1e-9f);
            if (iou > best) { best = iou; bi = g; }
        }

        const bool  pos  = best >= 0.5f;
        const bool  neg  = best <  0.4f;
        const float posf = pos ? 1.0f : 0.0f;

        const float x = p[4];
        const float obj_all = fmaxf(x, 0.f) - x * posf + log1pf(expf(-fabsf(x)));

        uint32_t key = 0u;
        if (neg) {  // softplus(x) > 0 -> bits are order-preserving as uint
            uint32_t kb = __float_as_uint(obj_all);
            key = kb ? kb : 1u;
            ++t_neg;
        }
        keys[(size_t)b * (size_t)(HW * 3) + n] = key;

        if (pos) {
            const float ax = (ax1 + ax2) * 0.5f, ay = (ay1 + ay2) * 0.5f;
            const float aw = fmaxf(ax2 - ax1, F_EPS), ah = fmaxf(ay2 - ay1, F_EPS);
            const float ttx = (s_gx[bi] - ax) / aw;
            const float tty = (s_gy[bi] - ay) / ah;
            const float ttw = logf(s_gw[bi] / aw);
            const float tth = logf(s_gh[bi] / ah);
            t_loc += sl1(p[0] - ttx) + sl1(p[1] - tty) + sl1(p[2] - ttw) + sl1(p[3] - tth);
            t_obj += obj_all;
            const float m   = fmaxf(p[5], fmaxf(p[6], p[7]));
            const float lse = m + logf(expf(p[5] - m) + expf(p[6] - m) + expf(p[7] - m));
            int tgt = s_lab[bi]; if (tgt < 0) tgt = 0;
            const float pl = (tgt == 0) ? p[5] : ((tgt == 1) ? p[6] : p[7]);
            t_cls += lse - pl;
            ++t_pos;
        }
    }

    t_loc = waveSumF(t_loc); t_obj = waveSumF(t_obj); t_cls = waveSumF(t_cls);
    t_pos = waveSumI(t_pos); t_neg = waveSumI(t_neg);
    if (lane == 0) {
        s_redF[0][wave] = t_loc; s_redF[1][wave] = t_obj; s_redF[2][wave] = t_cls;
        s_redI[0][wave] = t_pos; s_redI[1][wave] = t_neg;
    }
    __syncthreads();
    if (tid == 0) {
        float L = 0, O = 0, Cc = 0; int P = 0, Ng = 0;
        #pragma unroll
        for (int w = 0; w < 8; ++w) {
            L += s_redF[0][w]; O += s_redF[1][w]; Cc += s_redF[2][w];
            P += s_redI[0][w]; Ng += s_redI[1][w];
        }
        const int idx = scale * B_IMG + b;
        atomicAdd(&partF[idx * 3 + 0], L);
        atomicAdd(&partF[idx * 3 + 1], O);
        atomicAdd(&partF[idx * 3 + 2], Cc);
        atomicAdd(&partI[idx * 2 + 0], P);
        atomicAdd(&partI[idx * 2 + 1], Ng);
    }
}

// --------------------------- pass B ----------------------------------------
// One block per (scale, image): exact top-k sum via 4x 8-bit radix select.
__launch_bounds__(WGSZ)
__global__ void dl_pass_b(const uint32_t* __restrict__ keys,
                          const float* __restrict__ partF,
                          const int*   __restrict__ partI,
                          float* __restrict__ comb) {
    __shared__ int   hist[256];
    __shared__ int   sh_digit, sh_kr;
    __shared__ float s_ws[8];

    const int idx   = blockIdx.x;       // 0..191
    const int scale = idx >> 6;
    const int b     = idx & 63;
    const int    Ns[3]    = {49152, 12288, 3072};
    const size_t baseK[3] = {0, (size_t)64 * 49152, (size_t)64 * 49152 + (size_t)64 * 12288};
    const int N = Ns[scale];
    const uint32_t* k = keys + baseK[scale] + (size_t)b * (size_t)N;
    const int tid = threadIdx.x;

    const int posc = partI[idx * 2 + 0];
    const int negc = partI[idx * 2 + 1];
    long kneed = 3L * (long)(posc > 1 ? posc : 1);
    int  keff  = (int)(((long)negc < kneed) ? (long)negc : kneed);

    float    selSum = 0.0f;
    uint32_t prefix = 0u, pmask = 0u;
    int      kr     = keff;

    if (keff > 0) {
        for (int pass = 0; pass < 4; ++pass) {
            const int shift = 24 - 8 * pass;
            for (int i = tid; i < 256; i += WGSZ) hist[i] = 0;
            __syncthreads();
            for (int i = tid; i < N; i += WGSZ) {
                uint32_t v = k[i];
                if (v != 0u && (v & pmask) == prefix)
                    atomicAdd(&hist[(v >> shift) & 0xFFu], 1);
            }
            __syncthreads();
            if (tid == 0) {
                int acc = 0, dsel = 0, krem = kr;
                for (int d = 255; d >= 0; --d) {
                    int c = hist[d];
                    if (acc + c >= kr) { dsel = d; krem = kr - acc; break; }
                    acc += c;
                }
                sh_digit = dsel; sh_kr = krem;
            }
            __syncthreads();
            prefix |= ((uint32_t)sh_digit) << shift;
            pmask  |= 0xFFu << shift;
            kr = sh_kr;
            __syncthreads();
        }
        // prefix = exact k-th largest key; kr = #entries needed at that value.
        float s = 0.0f;
        for (int i = tid; i < N; i += WGSZ) {
            uint32_t v = k[i];
            if (v > prefix) s += __uint_as_float(v);
        }
        s = waveSumF(s);
        if ((tid & 31) == 0) s_ws[tid >> 5] = s;
        __syncthreads();
        if (tid == 0) {
            float tot = 0.0f;
            #pragma unroll
            for (int w = 0; w < 8; ++w) tot += s_ws[w];
            selSum = tot + (float)kr * __uint_as_float(prefix);
        }
    }
    if (tid == 0)
        comb[idx] = partF[idx * 3 + 0] + partF[idx * 3 + 1] + partF[idx * 3 + 2] + selSum;
}

// --------------------------- final reduce (WMMA) ----------------------------
__launch_bounds__(32)
__global__ void dl_final(const float* __restrict__ comb, float* __restrict__ out) {
    const int lane = threadIdx.x;
    float acc = 0.0f;
    #pragma unroll
    for (int i = lane; i < 192; i += 32) acc += comb[i];
#if HAVE_WMMA_F32
    // A(16x4): vgpr0 = acc (K=0 lanes0-15 / K=2 lanes16-31), vgpr1 = 0.
    // B(4x16) = ones  =>  D[m][n] = acc[m] + acc[m+16] for every n.
    v2f A;  A[0] = acc;  A[1] = 0.0f;
    v2f Bo; Bo[0] = 1.0f; Bo[1] = 1.0f;
    v8f C = {0.f, 0.f, 0.f, 0.f, 0.f, 0.f, 0.f, 0.f};
    C = __builtin_amdgcn_wmma_f32_16x16x4_f32(false, A, false, Bo, (short)0, C,
                                              false, false);
    float s = C[0] + C[1] + C[2] + C[3] + C[4] + C[5] + C[6] + C[7];
    s += __shfl_xor(s, 16, 32);   // halves hold M=0..7 / M=8..15 row sums
#else
    float s = waveSumF(acc);
#endif
    if (lane == 0) out[0] = s * (1.0f / 64.0f);
}

// --------------------------- launch -----------------------------------------
extern "C" void kernel_launch(void* const* d_in, const int* in_sizes, int n_in,
                              void* d_out, int out_size, void* d_ws, size_t ws_size,
                              hipStream_t stream) {
    (void)in_sizes; (void)n_in; (void)out_size; (void)ws_size;
    const float* pred0 = (const float*)d_in[0];
    const float* anch0 = (const float*)d_in[1];
    const float* pred1 = (const float*)d_in[2];
    const float* anch1 = (const float*)d_in[3];
    const float* pred2 = (const float*)d_in[4];
    const float* anch2 = (const float*)d_in[5];
    const float* gtb   = (const float*)d_in[6];
    const int*   gtl   = (const int*)d_in[7];

    char*     ws    = (char*)d_ws;
    float*    partF = (float*)(ws);              // 192*3 f32
    int*      partI = (int*)(ws + 2304);         // 192*2 i32
    float*    comb  = (float*)(ws + 3840);       // 192 f32
    uint32_t* keys  = (uint32_t*)(ws + 8192);    // 64*(49152+12288+3072) u32 ~ 16.5 MB

    dl_init<<<1, WGSZ, 0, stream>>>(partF, partI);

    dl_pass_a<<<dim3(16384 / WGSZ, B_IMG), WGSZ, 0, stream>>>(
        pred0, anch0, gtb, gtl, 16384, keys, partF, partI, 0);
    dl_pass_a<<<dim3(4096 / WGSZ, B_IMG), WGSZ, 0, stream>>>(
        pred1, anch1, gtb, gtl, 4096, keys + (size_t)64 * 49152, partF, partI, 1);
    dl_pass_a<<<dim3(1024 / WGSZ, B_IMG), WGSZ, 0, stream>>>(
        pred2, anch2, gtb, gtl, 1024,
        keys + (size_t)64 * 49152 + (size_t)64 * 12288, partF, partI, 2);

    dl_pass_b<<<192, WGSZ, 0, stream>>>(keys, partF, partI, comb);
    dl_final<<<1, 32, 0, stream>>>(comb, (float*)d_out);
}